// HiggsAudioModel_51384988729474
// MI455X (gfx1250) — compile-verified
//
#include <hip/hip_runtime.h>
#include <stdint.h>

#define T_IN_IDX  128011
#define T_OUT_IDX 128012
#define T_PAD_ID  128001
#define T_IGNORE  (-100)

#define AS_GLOBAL __attribute__((address_space(1)))
#define AS_LDS    __attribute__((address_space(3)))

typedef int v4i __attribute__((ext_vector_type(4)));
typedef AS_GLOBAL v4i* gv4i_p;   // global-AS pointer to 128-bit vector
typedef AS_LDS    v4i* lv4i_p;   // LDS-AS pointer to 128-bit vector

// CDNA5 async global<->LDS streaming path (ASYNCcnt-tracked). Existence of the
// builtins in this toolchain was proven by the round-1 type diagnostic.
#if defined(__HIP_DEVICE_COMPILE__)
#if __has_builtin(__builtin_amdgcn_global_load_async_to_lds_b128) && \
    __has_builtin(__builtin_amdgcn_global_store_async_from_lds_b128)
#define USE_ASYNC_LDS 1
#endif
#endif

__device__ __forceinline__ void wait_async0() {
#if defined(__HIP_DEVICE_COMPILE__)
#if __has_builtin(__builtin_amdgcn_s_wait_asynccnt)
  __builtin_amdgcn_s_wait_asynccnt(0);
#else
  asm volatile("s_wait_asynccnt 0" ::: "memory");
#endif
#endif
}

// Inclusive block scan over 256 per-thread values (Hillis-Steele in LDS).
__device__ __forceinline__ int blockScanIncl(int v, int* sbuf, int t, int* total) {
  sbuf[t] = v;
  __syncthreads();
  for (int off = 1; off < 256; off <<= 1) {
    int x = (t >= off) ? sbuf[t - off] : 0;
    __syncthreads();
    sbuf[t] += x;
    __syncthreads();
  }
  int incl = sbuf[t];
  *total = sbuf[255];
  __syncthreads();  // safe reuse of sbuf by caller
  return incl;
}

// ---------------------------------------------------------------------------
// K1: per-row placeholder counts + audio segment lengths (_codes_length)
// ---------------------------------------------------------------------------
__global__ void k_count(const int* __restrict__ input_ids,
                        const int* __restrict__ in_starts, const int* __restrict__ out_starts,
                        int* __restrict__ row_in_cnt, int* __restrict__ row_out_cnt,
                        int* __restrict__ in_len, int* __restrict__ out_len,
                        int S, int nIn, int nOut, int total_in, int total_out) {
  const int b = blockIdx.x, t = threadIdx.x;
  __shared__ int sbuf[256];
  int cin = 0, cout = 0;
  for (int s = t; s < S; s += 256) {
    const int id = input_ids[(size_t)b * S + s];
    cin += (id == T_IN_IDX);
    cout += (id == T_OUT_IDX);
  }
  int totIn, totOut;
  blockScanIncl(cin, sbuf, t, &totIn);
  blockScanIncl(cout, sbuf, t, &totOut);
  if (t == 0) { row_in_cnt[b] = totIn; row_out_cnt[b] = totOut; }
  if (b == 0) {
    if (t < nIn)  in_len[t]  = ((t + 1 < nIn)  ? in_starts[t + 1]  : total_in)  - in_starts[t];
    if (t < nOut) out_len[t] = ((t + 1 < nOut) ? out_starts[t + 1] : total_out) - out_starts[t];
  }
}

// ---------------------------------------------------------------------------
// K2: build inverse map src_code[b][m] = (type<<28)|src_index
//     type 0=pad, 1=text(idx=b*S+s), 2=audio-in(idx=embed row), 3=audio-out
// ---------------------------------------------------------------------------
__global__ void k_map(const int* __restrict__ input_ids,
                      const int* __restrict__ in_starts, const int* __restrict__ in_len,
                      const int* __restrict__ out_starts, const int* __restrict__ out_len,
                      const int* __restrict__ row_in_cnt, const int* __restrict__ row_out_cnt,
                      int* __restrict__ src_code, int S, int M) {
  const int b = blockIdx.x, t = threadIdx.x;
  __shared__ int sbuf[256];
  const int CH = (S + 255) >> 8;  // contiguous chunk per thread (<=16)
  int* row = src_code + (size_t)b * M;
  for (int m = t; m < M; m += 256) row[m] = 0;  // default: padding

  // row-major global placeholder ordinal base for this row (B<=8, cheap loop)
  int in_base = 0, out_base = 0;
  for (int r = 0; r < b; ++r) { in_base += row_in_cnt[r]; out_base += row_out_cnt[r]; }

  int ids[16];
  int cin = 0, cout = 0;
  for (int i = 0; i < CH; ++i) {
    const int s = t * CH + i;
    const int id = (s < S) ? input_ids[(size_t)b * S + s] : 0;
    ids[i] = id;
    cin += (id == T_IN_IDX);
    cout += (id == T_OUT_IDX);
  }
  int totIn, totOut, totTpn;
  const int inExcl  = blockScanIncl(cin,  sbuf, t, &totIn)  - cin;
  const int outExcl = blockScanIncl(cout, sbuf, t, &totOut) - cout;

  // tokens-per-position: 1 for text, segment length for placeholders
  int tpn[16];
  int tsum = 0;
  {
    int li = in_base + inExcl, lo = out_base + outExcl;
    for (int i = 0; i < CH; ++i) {
      const int s = t * CH + i;
      int v = 0;
      if (s < S) {
        const int id = ids[i];
        if (id == T_IN_IDX)       v = in_len[li++];
        else if (id == T_OUT_IDX) v = out_len[lo++];
        else                      v = 1;
      }
      tpn[i] = v;
      tsum += v;
    }
  }
  const int tExcl = blockScanIncl(tsum, sbuf, t, &totTpn) - tsum;
  const int shift = M - totTpn;  // left-padding shift (new_pos adjustment)

  // Phase 1: audio-in fills (matches reference fill order)
  {
    int run = tExcl, li = in_base + inExcl;
    for (int i = 0; i < CH; ++i) {
      const int s = t * CH + i;
      if (s >= S) break;
      run += tpn[i];
      if (ids[i] == T_IN_IDX) {
        const int p = li++;
        const int len = in_len[p], st = in_starts[p];
        const int c0 = (run - 1 + shift) - len + 1;
        for (int k = 0; k < len; ++k)
          row[c0 + k] = (int)((2u << 28) | (unsigned)(st + k));
      }
    }
  }
  __syncthreads();
  // Phase 2: audio-out fills
  {
    int run = tExcl, lo = out_base + outExcl;
    for (int i = 0; i < CH; ++i) {
      const int s = t * CH + i;
      if (s >= S) break;
      run += tpn[i];
      if (ids[i] == T_OUT_IDX) {
        const int p = lo++;
        const int len = out_len[p], st = out_starts[p];
        const int c0 = (run - 1 + shift) - len + 1;
        for (int k = 0; k < len; ++k)
          row[c0 + k] = (int)((3u << 28) | (unsigned)(st + k));
      }
    }
  }
  __syncthreads();
  // Phase 3: text scatter (last writer, like the reference)
  {
    int run = tExcl;
    for (int i = 0; i < CH; ++i) {
      const int s = t * CH + i;
      if (s >= S) break;
      run += tpn[i];
      const int id = ids[i];
      if (id != T_IN_IDX && id != T_OUT_IDX)
        row[run - 1 + shift] = (int)((1u << 28) | (unsigned)(b * S + s));
    }
  }
}

// ---------------------------------------------------------------------------
// K3: metadata outputs + attention cumsum -> position_ids (per row block scan)
//     meta layout: [att | labels | pos | input_ids | in_fill | in_discrete | out_fill]
// ---------------------------------------------------------------------------
__global__ void k_meta(const int* __restrict__ src_code, const int* __restrict__ input_ids,
                       const int* __restrict__ attn, const int* __restrict__ labels,
                       float* __restrict__ meta, unsigned long long BM, int S, int M) {
  const int b = blockIdx.x, t = threadIdx.x;
  __shared__ int sbuf[256];
  const int CH = (M + 255) >> 8;  // <=16 for M=4088
  const int* row = src_code + (size_t)b * M;
  int codes[16], att[16];
  int asum = 0;
  for (int i = 0; i < CH; ++i) {
    const int m = t * CH + i;
    const int code = (m < M) ? row[m] : 0;
    codes[i] = code;
    const unsigned ty = ((unsigned)code) >> 28;
    const int idx = code & 0x0FFFFFFF;
    int a = 0;
    if (m < M) {
      if (ty == 1u)      a = attn[idx];
      else if (ty >= 2u) a = 1;  // fills force attention on
    }
    att[i] = a;
    asum += a;
  }
  int tot;
  int run = blockScanIncl(asum, sbuf, t, &tot) - asum;
  for (int i = 0; i < CH; ++i) {
    const int m = t * CH + i;
    if (m >= M) break;
    const int code = codes[i];
    const unsigned ty = ((unsigned)code) >> 28;
    const int idx = code & 0x0FFFFFFF;
    run += att[i];
    float fid, flab;
    if (ty == 1u)      { fid = (float)input_ids[idx]; flab = (float)labels[idx]; }
    else if (ty == 2u) { fid = (float)T_IN_IDX;  flab = (float)T_IGNORE; }
    else if (ty == 3u) { fid = (float)T_OUT_IDX; flab = (float)T_IGNORE; }
    else               { fid = (float)T_PAD_ID;  flab = (float)T_IGNORE; }
    const float fpos = (att[i] == 0) ? 1.0f : (float)(run - 1);
    const unsigned long long o = (unsigned long long)b * M + m;
    const float fin = (ty == 2u) ? 1.0f : 0.0f;
    meta[o]            = (float)att[i];               // final_attention
    meta[BM + o]       = flab;                        // final_labels
    meta[2 * BM + o]   = fpos;                        // position_ids
    meta[3 * BM + o]   = fid;                         // final_input_ids
    meta[4 * BM + o]   = fin;                         // in_fill
    meta[5 * BM + o]   = fin;                         // in_discrete
    meta[6 * BM + o]   = (ty == 3u) ? 1.0f : 0.0f;    // out_fill
  }
}

// ---------------------------------------------------------------------------
// K4: embedding gather — the bandwidth-dominant kernel (~0.5 GB of traffic).
//     One block per merged column; 8 KB row staged via CDNA5 async-LDS b128.
// ---------------------------------------------------------------------------
__global__ void k_embed(const float* __restrict__ a_in, const float* __restrict__ a_out,
                        const float* __restrict__ t_emb, const int* __restrict__ src_code,
                        float* __restrict__ dst, int D) {
  const int col = blockIdx.x;  // flattened b*M + m
  const int t = threadIdx.x;
  const int code = src_code[col];
  const unsigned ty = ((unsigned)code) >> 28;
  const int idx = code & 0x0FFFFFFF;
  float* out = dst + (size_t)col * (size_t)D;
  if (ty == 0u) {  // padding column -> zeros
    const float4 z = make_float4(0.0f, 0.0f, 0.0f, 0.0f);
    for (int e = t * 4; e + 3 < D; e += 1024) *(float4*)(out + e) = z;
    return;
  }
  const float* src = (ty == 1u) ? (t_emb + (size_t)idx * D)
                   : (ty == 2u) ? (a_in + (size_t)idx * D)
                                : (a_out + (size_t)idx * D);
#if defined(USE_ASYNC_LDS)
  if (D <= 2048 && (D & 3) == 0) {
    // 16B-aligned staging buffer: async b128 ops move 16-byte quanta and LDS
    // B128 wants native 16B alignment; lane offsets (t*4 floats) are 16B steps.
    __shared__ __align__(16) float lds[2048];
    // global -> LDS, lane-private 16B chunks, tracked on ASYNCcnt
    for (int e = t * 4; e < D; e += 1024)
      __builtin_amdgcn_global_load_async_to_lds_b128(
          (gv4i_p)(src + e), (lv4i_p)(lds + e), 0, 0);
    wait_async0();
    // LDS -> global (same lanes consume their own chunks: no barrier needed)
    for (int e = t * 4; e < D; e += 1024)
      __builtin_amdgcn_global_store_async_from_lds_b128(
          (gv4i_p)(out + e), (lv4i_p)(lds + e), 0, 0);
    wait_async0();
    return;
  }
#endif
  for (int e = t * 4; e + 3 < D; e += 1024)
    *(float4*)(out + e) = *(const float4*)(src + e);
}

// ---------------------------------------------------------------------------
extern "C" void kernel_launch(void* const* d_in, const int* in_sizes, int n_in,
                              void* d_out, int out_size, void* d_ws, size_t ws_size,
                              hipStream_t stream) {
  const float* a_in_emb  = (const float*)d_in[0];   // (n_in*L, D)
  const float* a_out_emb = (const float*)d_in[1];   // (n_out*L, D)
  const float* t_emb     = (const float*)d_in[2];   // (B, S, D)
  const int* in_starts   = (const int*)d_in[3];
  const int* out_starts  = (const int*)d_in[4];
  const int* input_ids   = (const int*)d_in[5];
  const int* attn_mask   = (const int*)d_in[6];
  const int* label_ids   = (const int*)d_in[7];

  const int nIn  = in_sizes[3];
  const int nOut = in_sizes[4];
  const int BS   = in_sizes[5];
  const int B    = 8;                       // fixed by the reference problem
  const int S    = BS / B;
  const int D    = in_sizes[2] / BS;
  const int M    = (int)((long long)out_size / ((long long)B * (D + 7)));
  const int total_in  = in_sizes[0] / D;
  const int total_out = in_sizes[1] / D;

  // workspace carve-out (256B aligned slots)
  char* w = (char*)d_ws;
  int* row_in_cnt  = (int*)w; w += 256;
  int* row_out_cnt = (int*)w; w += 256;
  int* in_len      = (int*)w; w += ((nIn  * 4 + 255) / 256) * 256;
  int* out_len     = (int*)w; w += ((nOut * 4 + 255) / 256) * 256;
  int* src_code    = (int*)w;  // B*M ints (~128 KB)

  const unsigned long long BM = (unsigned long long)B * (unsigned long long)M;
  float* out = (float*)d_out;
  float* meta = out + BM * (unsigned long long)D;

  k_count<<<B, 256, 0, stream>>>(input_ids, in_starts, out_starts,
                                 row_in_cnt, row_out_cnt, in_len, out_len,
                                 S, nIn, nOut, total_in, total_out);
  k_map<<<B, 256, 0, stream>>>(input_ids, in_starts, in_len, out_starts, out_len,
                               row_in_cnt, row_out_cnt, src_code, S, M);
  k_meta<<<B, 256, 0, stream>>>(src_code, input_ids, attn_mask, label_ids,
                                meta, BM, S, M);
  k_embed<<<(unsigned)BM, 256, 0, stream>>>(a_in_emb, a_out_emb, t_emb,
                                            src_code, out, D);
}